// BoxE_19370302505588
// MI455X (gfx1250) — compile-verified
//
#include <hip/hip_runtime.h>
#include <math.h>

#define EMB      256
#define RELW     (4 * EMB + 2)   /* 1026 floats per relation row */
#define EPS_F    1e-6f

/* ---- fast HW math (guarded: fall back to libm if builtin absent) ---- */
#if __has_builtin(__builtin_amdgcn_tanhf)
#define TANHF(x) __builtin_amdgcn_tanhf(x)
#else
#define TANHF(x) tanhf(x)
#endif
#if __has_builtin(__builtin_amdgcn_logf)          /* v_log_f32: log2 */
#define LOG2F(x) __builtin_amdgcn_logf(x)
#else
#define LOG2F(x) __log2f(x)
#endif
#if __has_builtin(__builtin_amdgcn_exp2f)         /* v_exp_f32: exp2 */
#define EXP2F(x) __builtin_amdgcn_exp2f(x)
#else
#define EXP2F(x) exp2f(x)
#endif
#if __has_builtin(__builtin_amdgcn_rcpf)
#define RCPF(x)  __builtin_amdgcn_rcpf(x)
#else
#define RCPF(x)  (1.0f / (x))
#endif
#if __has_builtin(__builtin_amdgcn_sqrtf)
#define SQRTF(x) __builtin_amdgcn_sqrtf(x)
#else
#define SQRTF(x) sqrtf(x)
#endif

/* ---- CDNA5 async global->LDS copy (ASYNCcnt-tracked) ----
 * INST_OFFSET is added to BOTH the LDS and the global address (ISA 08 §4.4),
 * so one (ldsaddr, voff) pair covers a whole row with immediate offsets.
 * th:TH_LOAD_NT: head/tail are streamed once -> keep them non-temporal in L2
 * so the randomly-gathered 4.2 MB relation table stays L2-resident. */
#define ASYNC_LD_B128_NT(ldsaddr, voff, sbase, imm)                           \
  asm volatile("global_load_async_to_lds_b128 %0, %1, %2 offset:" #imm        \
               " th:TH_LOAD_NT"                                               \
               :: "v"(ldsaddr), "v"(voff), "s"(sbase) : "memory")

#if __has_builtin(__builtin_amdgcn_s_wait_asynccnt)
#define WAIT_ASYNCCNT(n) __builtin_amdgcn_s_wait_asynccnt(n)
#else
#define WAIT_ASYNCCNT(n) asm volatile("s_wait_asynccnt " #n ::: "memory")
#endif

__device__ __forceinline__ float waveRedAdd32(float v) {
  #pragma unroll
  for (int m = 16; m > 0; m >>= 1) v += __shfl_xor(v, m, 32);
  return v;
}

/* 8 floats per lane from LDS: float4 at [lane] and [32+lane] (16B aligned) */
__device__ __forceinline__ void load8_lds(float* dst, const float* base, int lane) {
  const float4* p = (const float4*)base;
  float4 a = p[lane];
  float4 b = p[32 + lane];
  dst[0] = a.x; dst[1] = a.y; dst[2] = a.z; dst[3] = a.w;
  dst[4] = b.x; dst[5] = b.y; dst[6] = b.z; dst[7] = b.w;
}

/* 8 floats per lane from the relation row (only 8B-aligned: 1026-float stride) */
__device__ __forceinline__ void load8_rel(float* dst, const float* base, int lane) {
  const float2* p = (const float2*)base;
  float2 a0 = p[2 * lane],      a1 = p[2 * lane + 1];
  float2 b0 = p[64 + 2 * lane], b1 = p[64 + 2 * lane + 1];
  dst[0] = a0.x; dst[1] = a0.y; dst[2] = a1.x; dst[3] = a1.y;
  dst[4] = b0.x; dst[5] = b0.y; dst[6] = b1.x; dst[7] = b1.y;
}

/* one (row, ht) slab: returns L2 norm of dist over the 256 dims */
__device__ __forceinline__ float score_ht(const float* wraw, const float* cen,
                                          const float* bmp, float bs) {
  float slw = 0.0f;
  #pragma unroll
  for (int e = 0; e < 8; ++e) slw += LOG2F(fmaxf(fabsf(wraw[e]), EPS_F));
  float tot   = waveRedAdd32(slw);
  float gmean = EXP2F(tot * (1.0f / 256.0f));                 /* exp(mean(ln w)) */
  float elu1  = (bs > 0.0f) ? (1.0f + bs) : EXP2F(bs * 1.44269504f);
  float f     = elu1 / fmaxf(gmean, EPS_F);

  float acc = 0.0f;
  #pragma unroll
  for (int e = 0; e < 8; ++e) {
    float wi   = TANHF(fabsf(wraw[e]) * f);
    float ci   = TANHF(cen[e]);
    float bi   = TANHF(bmp[e]);
    float wp1  = wi + 1.0f;
    float inv  = RCPF(wp1);
    float cd   = fabsf(bi - ci);
    float kap  = 0.5f * wi * (wp1 - inv);
    float dist = (cd <= 0.5f * wi) ? (cd * inv) : fmaf(cd, wp1, -kap);
    acc = fmaf(dist, dist, acc);
  }
  return SQRTF(waveRedAdd32(acc));
}

__global__ __launch_bounds__(256) void boxe_score_kernel(
    const float* __restrict__ head, const int* __restrict__ relid,
    const float* __restrict__ tail, const float* __restrict__ rel,
    float* __restrict__ out, int nrows, int rows_per_wave) {
  /* 8 waves * 2 buffers * 1024 floats (head row 512 | tail row 512) = 64 KB */
  __shared__ __align__(16) float smem[8][2][1024];

  const int lane  = threadIdx.x & 31;
  const int wave  = threadIdx.x >> 5;
  const int gwave = blockIdx.x * 8 + wave;
  const long row0 = (long)gwave * rows_per_wave;
  if (row0 >= nrows) return;
  int nmine = rows_per_wave;
  if (row0 + nmine > nrows) nmine = (int)(nrows - row0);

  const unsigned ldsH = (unsigned)(size_t)(&smem[wave][0][0])   + (unsigned)lane * 16u;
  const unsigned ldsT = (unsigned)(size_t)(&smem[wave][0][512]) + (unsigned)lane * 16u;

  auto issue = [&](long row, int buf) {
    unsigned lh  = ldsH + (unsigned)buf * 4096u;
    unsigned lt  = ldsT + (unsigned)buf * 4096u;
    unsigned off = (unsigned)(row * 2048L) + (unsigned)lane * 16u; /* bytes */
    ASYNC_LD_B128_NT(lh, off, head, 0);
    ASYNC_LD_B128_NT(lh, off, head, 512);
    ASYNC_LD_B128_NT(lh, off, head, 1024);
    ASYNC_LD_B128_NT(lh, off, head, 1536);
    ASYNC_LD_B128_NT(lt, off, tail, 0);
    ASYNC_LD_B128_NT(lt, off, tail, 512);
    ASYNC_LD_B128_NT(lt, off, tail, 1024);
    ASYNC_LD_B128_NT(lt, off, tail, 1536);
  };

  issue(row0, 0);                                  /* prologue: fill buffer 0 */

  for (int i = 0; i < nmine; ++i) {
    const long row = row0 + i;
    const int  buf = i & 1;
    const bool more = (i + 1 < nmine);
    if (more) {
      issue(row + 1, buf ^ 1);                     /* prefetch next row's DMA */
      int nrid = relid[row + 1];                   /* warm next relation row  */
      __builtin_prefetch((const char*)(rel + (long)nrid * RELW) + lane * 128, 0, 3);
    }

    /* relation row loads (L2-resident) issued BEFORE the async wait */
    const int rid = relid[row];
    const float* rrow = rel + (long)rid * RELW;
    float C0[8], C1[8], W0[8], W1[8];
    load8_rel(C0, rrow,            lane);
    load8_rel(C1, rrow + EMB,      lane);
    load8_rel(W0, rrow + 2 * EMB,  lane);
    load8_rel(W1, rrow + 3 * EMB,  lane);
    const float bs0 = rrow[4 * EMB], bs1 = rrow[4 * EMB + 1];

    /* wait: current buffer's 8 DMAs done; next row's 8 may stay in flight */
    if (more) { WAIT_ASYNCCNT(8); } else { WAIT_ASYNCCNT(0); }
    asm volatile("" ::: "memory");

    const float* hrow = &smem[wave][buf][0];
    const float* trow = &smem[wave][buf][512];
    float HP[8], HB[8], TP[8], TB[8];
    load8_lds(HP, hrow,       lane);
    load8_lds(HB, hrow + EMB, lane);
    load8_lds(TP, trow,       lane);
    load8_lds(TB, trow + EMB, lane);

    float B0[8], B1[8];
    #pragma unroll
    for (int e = 0; e < 8; ++e) { B0[e] = HP[e] + TB[e]; B1[e] = TP[e] + HB[e]; }

    const float n0 = score_ht(W0, C0, B0, bs0);
    const float n1 = score_ht(W1, C1, B1, bs1);
    if (lane == 0) __builtin_nontemporal_store(-(n0 + n1), &out[row]);
  }
}

extern "C" void kernel_launch(void* const* d_in, const int* in_sizes, int n_in,
                              void* d_out, int out_size, void* d_ws, size_t ws_size,
                              hipStream_t stream) {
  const float* head  = (const float*)d_in[0];
  const int*   relid = (const int*)d_in[1];
  const float* tail  = (const float*)d_in[2];
  const float* rel   = (const float*)d_in[3];
  float*       out   = (float*)d_out;

  const int nrows = in_sizes[1];                 /* BATCH = 131072 */
  const int waves_per_block = 8;                 /* 256 threads, wave32 */
  const int target_waves = 16384;
  int rows_per_wave = (nrows + target_waves - 1) / target_waves;  /* 8 */
  if (rows_per_wave < 1) rows_per_wave = 1;
  const int rows_per_block = rows_per_wave * waves_per_block;
  const int blocks = (nrows + rows_per_block - 1) / rows_per_block;

  boxe_score_kernel<<<blocks, 256, 0, stream>>>(head, relid, tail, rel, out,
                                                nrows, rows_per_wave);
}